// MedianPool_26920855011346
// MI455X (gfx1250) — compile-verified
//
#include <hip/hip_runtime.h>
#include <stdint.h>

// 17x17 same-padded lower-median filter over (1,4,256,256) fp32.
// Per block: TDM-stage a clamped 32x32 halo tile into LDS (tensor_load_to_lds,
// TENSORcnt), transform fp32 bits to monotone u32 keys, then each thread does
// an exact 32-pass binary radix-select for one output pixel.
// Grid: 16x16 tiles x 4 channels = 1024 blocks of 256 threads (8 waves).

#define HW    256
#define TILE  16
#define REGN  32          // 16 + 2*8 halo
#define RAD   8
#define KMED  144         // 0-based index of lower median among 289

typedef unsigned int u32x4 __attribute__((ext_vector_type(4)));
typedef int          i32x4 __attribute__((ext_vector_type(4)));
typedef int          i32x8 __attribute__((ext_vector_type(8)));

// Monotone map: fp32 bits -> u32 preserving total order (-0 < +0).
__device__ __forceinline__ unsigned f2key(unsigned u) {
  return u ^ (unsigned)(((int)u >> 31) | (int)0x80000000);
}

__global__ __launch_bounds__(256)
void median17_kernel(const float* __restrict__ x, float* __restrict__ out) {
  __shared__ unsigned keys[REGN * REGN];   // 4096 B

  const int c   = blockIdx.z;
  const int ty0 = blockIdx.y * TILE;
  const int tx0 = blockIdx.x * TILE;

  // Clamped 32x32 staging region: always fully inside the 256x256 plane,
  // and always contains every in-bounds window pixel for this tile.
  int ry0 = ty0 - RAD; ry0 = ry0 < 0 ? 0 : (ry0 > HW - REGN ? HW - REGN : ry0);
  int rx0 = tx0 - RAD; rx0 = rx0 < 0 ? 0 : (rx0 > HW - REGN ? HW - REGN : rx0);

  const float* plane = x + (size_t)c * HW * HW;

  // ---------------- stage 32x32 tile into LDS ----------------
#if defined(__AMDGCN__) && __has_builtin(__builtin_amdgcn_tensor_load_to_lds)
  {
    // Only wave 0 issues the DMA. TDM ignores EXEC, so the guard must be
    // a *scalar* (wave-uniform) branch: readfirstlane makes that explicit.
    const int wave = __builtin_amdgcn_readfirstlane((int)(threadIdx.x >> 5));
    if (wave == 0) {
      unsigned long long ga =
          (unsigned long long)(uintptr_t)(plane + (size_t)ry0 * HW + rx0);
      const unsigned lds_addr = (unsigned)(uintptr_t)(&keys[0]); // low 32 = LDS byte offset

      u32x4 g0;
      g0[0] = 1u;                                  // count=1 (valid), no gather
      g0[1] = lds_addr;                            // D#.lds_addr
      g0[2] = (unsigned)ga;                        // global_addr[31:0]
      g0[3] = (unsigned)(ga >> 32) | (2u << 30);   // global_addr[56:32] | type=2

      i32x8 g1;
      g1[0] = (int)(2u << 16);                             // data_size = 4B
      g1[1] = (int)((HW & 0xFFFFu) << 16);                 // tensor_dim0 lo16
      g1[2] = (int)(((HW >> 16) & 0xFFFFu)
                    | ((HW & 0xFFFFu) << 16));             // dim0 hi16 | dim1 lo16
      g1[3] = (int)(((HW >> 16) & 0xFFFFu)
                    | ((unsigned)REGN << 16));             // dim1 hi16 | tile_dim0=32
      g1[4] = (int)REGN;                                   // tile_dim1=32, tile_dim2=0
      g1[5] = (int)HW;                                     // tensor_dim0_stride = 256
      g1[6] = 0;                                           // stride0 hi | stride1 lo
      g1[7] = 0;

      i32x4 gz = {0, 0, 0, 0};                             // 2-D: groups 2/3 unused
#if __clang_major__ >= 23
      i32x8 gz8 = {0, 0, 0, 0, 0, 0, 0, 0};
      __builtin_amdgcn_tensor_load_to_lds(g0, g1, gz, gz, gz8, 0);
#else
      __builtin_amdgcn_tensor_load_to_lds(g0, g1, gz, gz, 0);
#endif
#if __has_builtin(__builtin_amdgcn_s_wait_tensorcnt)
      __builtin_amdgcn_s_wait_tensorcnt(0);
#else
      asm volatile("s_wait_tensorcnt 0x0" ::: "memory");
#endif
    }
    __syncthreads();
    // In-place: raw fp32 bits -> monotone keys (four elements per thread, stride 256).
    for (int i = threadIdx.x; i < REGN * REGN; i += 256)
      keys[i] = f2key(keys[i]);
  }
#elif defined(__AMDGCN__) && __has_builtin(__builtin_amdgcn_global_load_async_to_lds_b32)
  for (int i = threadIdx.x; i < REGN * REGN; i += 256) {
    const int ry = i / REGN, rx = i - ry * REGN;
    __builtin_amdgcn_global_load_async_to_lds_b32(
        (const __attribute__((address_space(1))) void*)
            (plane + (size_t)(ry0 + ry) * HW + (rx0 + rx)),
        (__attribute__((address_space(3))) void*)&keys[i], 0, 0);
  }
#if __has_builtin(__builtin_amdgcn_s_wait_asynccnt)
  __builtin_amdgcn_s_wait_asynccnt(0);
#else
  asm volatile("s_wait_asynccnt 0x0" ::: "memory");
#endif
  __syncthreads();
  for (int i = threadIdx.x; i < REGN * REGN; i += 256)
    keys[i] = f2key(keys[i]);
#else
  for (int i = threadIdx.x; i < REGN * REGN; i += 256) {
    const int ry = i / REGN, rx = i - ry * REGN;
    keys[i] = f2key(__float_as_uint(plane[(size_t)(ry0 + ry) * HW + (rx0 + rx)]));
  }
#endif
  __syncthreads();

  // ---------------- per-pixel exact median (binary radix select) ----------
  const int tx = (int)(threadIdx.x & (TILE - 1));   // 0..15
  const int ty = (int)(threadIdx.x >> 4);           // 0..15

  const int oy = ty0 + ty;
  const int ox = tx0 + tx;

  // In-bounds part of this pixel's 17x17 window is a rectangle.
  int cy0 = oy - RAD; if (cy0 < 0) cy0 = 0;
  int cy1 = oy + RAD; if (cy1 > HW - 1) cy1 = HW - 1;
  int cx0 = ox - RAD; if (cx0 < 0) cx0 = 0;
  int cx1 = ox + RAD; if (cx1 > HW - 1) cx1 = HW - 1;
  const int h = cy1 - cy0 + 1;
  const int w = cx1 - cx0 + 1;
  const int noob = 17 * 17 - h * w;          // zero-padding count, key 0x80000000

  const unsigned* wbase = &keys[(cy0 - ry0) * REGN + (cx0 - rx0)];

  unsigned res = 0u;
  if (noob == 0) {
    // Interior fast path: fixed 17x17 bounds, inner row unrolled.
    for (int b = 31; b >= 0; --b) {
      const unsigned t = res | (1u << b);
      int cnt = 0;
      const unsigned* p = wbase;
      for (int yy = 0; yy < 17; ++yy) {
#pragma unroll
        for (int xx = 0; xx < 17; ++xx) cnt += (p[xx] < t) ? 1 : 0;
        p += REGN;
      }
      if (cnt <= KMED) res = t;
    }
  } else {
    for (int b = 31; b >= 0; --b) {
      const unsigned t = res | (1u << b);
      int cnt = (t > 0x80000000u) ? noob : 0;  // padding zeros, counted analytically
      const unsigned* p = wbase;
      for (int yy = 0; yy < h; ++yy) {
        for (int xx = 0; xx < w; ++xx) cnt += (p[xx] < t) ? 1 : 0;
        p += REGN;
      }
      if (cnt <= KMED) res = t;
    }
  }
  // Invert the monotone map.
  const unsigned u = (res & 0x80000000u) ? (res ^ 0x80000000u) : ~res;
  out[((size_t)c * HW + oy) * HW + ox] = __uint_as_float(u);
}

extern "C" void kernel_launch(void* const* d_in, const int* in_sizes, int n_in,
                              void* d_out, int out_size, void* d_ws, size_t ws_size,
                              hipStream_t stream) {
  (void)in_sizes; (void)n_in; (void)d_ws; (void)ws_size; (void)out_size;
  const float* x = (const float*)d_in[0];
  float* out = (float*)d_out;
  dim3 grid(HW / TILE, HW / TILE, 4);   // 16 x 16 tiles x 4 channels = 1024 blocks
  median17_kernel<<<grid, dim3(256, 1, 1), 0, stream>>>(x, out);
}